// Ive_38036230373446
// MI455X (gfx1250) — compile-verified
//
#include <hip/hip_runtime.h>

// ---------------------------------------------------------------------------
// i1e(z): exponentially scaled modified Bessel I1, single precision (Cephes
// i1ef algorithm — identical branch structure to scipy.special.i1e).
//   z <= 8 : i1e(z) = chbevl(z/2 - 2, A[17]) * z     (series is exp-scaled)
//   z >  8 : i1e(z) = chbevl(32/z - 2, B[7]) / sqrt(z)
// Math on float2 pairs -> packed v_pk_fma_f32 / v_pk_add_f32 (2 FP32/instr),
// plus per-element v_rcp_f32 / v_rsq_f32. No exp(), no division.
// Tail handled by overlap-clamping the last thread: every thread executes the
// same straight-line 8-element vector path (pure function -> overlapping
// writes are identical and deterministic).
// ---------------------------------------------------------------------------

typedef float v4f __attribute__((ext_vector_type(4)));
typedef float v2f __attribute__((ext_vector_type(2)));

__device__ __forceinline__ v2f splat2(float c) { v2f r = {c, c}; return r; }

__device__ __forceinline__ v2f chbevl_A17_v2(v2f x) {
    v2f b0 = splat2(9.38153738649577178388E-9f);
    v2f b1 = splat2(0.0f);
    v2f b2;
#define CHB_STEP(c) b2 = b1; b1 = b0; \
    b0 = __builtin_elementwise_fma(x, b1, splat2(c)) - b2;
    CHB_STEP(-4.44505912879632808065E-8f)
    CHB_STEP( 2.00329475355213526229E-7f)
    CHB_STEP(-8.56872026469545474066E-7f)
    CHB_STEP( 3.47025130813767847674E-6f)
    CHB_STEP(-1.32731636560394358279E-5f)
    CHB_STEP( 4.78156510755005422638E-5f)
    CHB_STEP(-1.61760815825896745588E-4f)
    CHB_STEP( 5.12285956168575772895E-4f)
    CHB_STEP(-1.51357245063125314899E-3f)
    CHB_STEP( 4.15642294431288815669E-3f)
    CHB_STEP(-1.05640848946261981558E-2f)
    CHB_STEP( 2.47264490306265168283E-2f)
    CHB_STEP(-5.29459812080949914269E-2f)
    CHB_STEP( 1.02643658689847095384E-1f)
    CHB_STEP(-1.76416518357834055153E-1f)
    CHB_STEP( 2.52587186443633654823E-1f)
    return 0.5f * (b0 - b2);
}

__device__ __forceinline__ v2f chbevl_B7_v2(v2f x) {
    v2f b0 = splat2(-3.83538038596423702205E-9f);
    v2f b1 = splat2(0.0f);
    v2f b2;
    CHB_STEP(-2.63146884688951950684E-8f)
    CHB_STEP(-2.51223623787020892529E-7f)
    CHB_STEP(-3.88256480887769039346E-6f)
    CHB_STEP(-1.10588938762623716291E-4f)
    CHB_STEP(-9.76109749136146840777E-3f)
    CHB_STEP( 7.78576235018280120474E-1f)
#undef CHB_STEP
    return 0.5f * (b0 - b2);
}

__device__ __forceinline__ v2f i1e_pair(v2f xx) {
    v2f z;
    z.x = fabsf(xx.x);
    z.y = fabsf(xx.y);

    // Small-z path: y in [-2,2]; packed Clenshaw
    v2f ys = __builtin_elementwise_fma(splat2(0.5f), z, splat2(-2.0f));
    v2f rs = chbevl_A17_v2(ys) * z;

    // Large-z path: 32/z - 2 via fast v_rcp_f32; scale by v_rsq_f32
    v2f rz, rq;
    rz.x = __builtin_amdgcn_rcpf(z.x);
    rz.y = __builtin_amdgcn_rcpf(z.y);
    rq.x = __builtin_amdgcn_rsqf(z.x);
    rq.y = __builtin_amdgcn_rsqf(z.y);
    v2f yl = __builtin_elementwise_fma(splat2(32.0f), rz, splat2(-2.0f));
    v2f rl = chbevl_B7_v2(yl) * rq;

    // Branchless selects (random data keeps both paths live in every wave)
    v2f r;
    r.x = (z.x <= 8.0f) ? rs.x : rl.x;
    r.y = (z.y <= 8.0f) ? rs.y : rl.y;
    r.x = (xx.x < 0.0f) ? -r.x : r.x;   // i1e is odd
    r.y = (xx.y < 0.0f) ? -r.y : r.y;
    return r;
}

// Hot kernel: requires n >= 8. Last threads clamp and overlap; no tail path.
__global__ void __launch_bounds__(256)
i1e_kernel_vec(const float* __restrict__ in, float* __restrict__ out, int n) {
    const int tid = blockIdx.x * blockDim.x + threadIdx.x;
    int base = tid * 8;
    const int last = n - 8;
    base = (base < last) ? base : last;   // overlap-clamp; branch-free

    const v4f* pin = reinterpret_cast<const v4f*>(in + base);
    v4f a = __builtin_nontemporal_load(pin + 0);
    v4f b = __builtin_nontemporal_load(pin + 1);

    v2f a_lo = {a.x, a.y}, a_hi = {a.z, a.w};
    v2f b_lo = {b.x, b.y}, b_hi = {b.z, b.w};

    v2f r0 = i1e_pair(a_lo);
    v2f r1 = i1e_pair(a_hi);
    v2f r2 = i1e_pair(b_lo);
    v2f r3 = i1e_pair(b_hi);

    v4f ra = {r0.x, r0.y, r1.x, r1.y};
    v4f rb = {r2.x, r2.y, r3.x, r3.y};

    v4f* pout = reinterpret_cast<v4f*>(out + base);
    __builtin_nontemporal_store(ra, pout + 0);
    __builtin_nontemporal_store(rb, pout + 1);
}

// Degenerate fallback for n < 8 (never taken for 16384x4096).
__global__ void i1e_kernel_small(const float* __restrict__ in,
                                 float* __restrict__ out, int n) {
    const int i = threadIdx.x;
    if (i < n) {
        v2f p = {in[i], in[i]};
        v2f r = i1e_pair(p);
        out[i] = r.x;
    }
}

extern "C" void kernel_launch(void* const* d_in, const int* in_sizes, int n_in,
                              void* d_out, int out_size, void* d_ws, size_t ws_size,
                              hipStream_t stream) {
    (void)n_in; (void)d_ws; (void)ws_size; (void)out_size;
    const float* z   = reinterpret_cast<const float*>(d_in[0]);
    float*       out = reinterpret_cast<float*>(d_out);
    const int n = in_sizes[0];            // 16384 * 4096 = 67,108,864

    if (n >= 8) {
        const int elems_per_thread = 8;
        const int threads = 256;
        const int total_threads = (n + elems_per_thread - 1) / elems_per_thread;
        const int blocks = (total_threads + threads - 1) / threads;
        i1e_kernel_vec<<<blocks, threads, 0, stream>>>(z, out, n);
    } else {
        i1e_kernel_small<<<1, 8, 0, stream>>>(z, out, n);
    }
}